// TorchLMHeadCE_64553358459088
// MI455X (gfx1250) — compile-verified
//
#include <hip/hip_runtime.h>

typedef float v2f __attribute__((ext_vector_type(2)));
typedef float v8f __attribute__((ext_vector_type(8)));

#define H_DIM     2048
#define V_DIM     128256
#define SOFTCAP   30.0f
#define ZLOSS     1e-4f
#define EPS_LS    0.1f
#define IGNORE_IDX (-100)

#define KC  32      // K chunk staged in LDS
#define MT  32      // rows per block
#define NT  256     // vocab cols per block (V/256 = 501 exactly)
#define LDP 36      // LDS row stride (floats): 16B-aligned rows, conflict-free b64 reads

// ---------------------------------------------------------------------------
// async global -> LDS (CDNA5): b128 per lane, tracked by ASYNCcnt
// lds: 32-bit LDS byte offset (low 32 bits of generic shared pointer)
// ---------------------------------------------------------------------------
__device__ __forceinline__ void async_b128(unsigned lds, const float* g) {
    asm volatile("global_load_async_to_lds_b128 %0, %1, off"
                 :: "v"(lds), "v"(g) : "memory");
}

// ---------------------------------------------------------------------------
// Kernel 0: zero workspace partials
// ---------------------------------------------------------------------------
__global__ void zero_ws(float* p, int n) {
    int i = blockIdx.x * blockDim.x + threadIdx.x;
    if (i < n) p[i] = 0.0f;
}

// ---------------------------------------------------------------------------
// Kernel 1: fused GEMM tile + softcap + partial reductions
//   S1[n] += sum_j exp(softcap(z_nj) - 30)   (fixed max = 30: softcap bounds z)
//   S2[n] += sum_j softcap(z_nj)
//   zy[n]  = softcap(z_{n, y[n]})            (unique writer across whole grid)
// ---------------------------------------------------------------------------
__global__ __launch_bounds__(256)
void lmhead_partial(const float* __restrict__ x, const float* __restrict__ W,
                    const int* __restrict__ y,
                    float* __restrict__ S1, float* __restrict__ S2,
                    float* __restrict__ zy) {
    __shared__ __align__(16) float xs [2][MT][LDP];   // 9.2 KB
    __shared__ __align__(16) float wsh[2][NT][LDP];   // 73.7 KB

    const int tid     = threadIdx.x;
    const int rowBase = blockIdx.x * MT;      // rows fastest -> W tile reuse in L2
    const int vBase   = blockIdx.y * NT;

    // loader mapping: x tile 32x32 -> 1 float4/thread; W tile 256x32 -> 1 row/thread
    const int xr = tid >> 3, xc = (tid & 7) << 2;
    const float* xg = x + (size_t)(rowBase + xr) * H_DIM + xc;
    const float* wg = W + (size_t)(vBase + tid) * H_DIM;

    const unsigned xlds0 = (unsigned)(size_t)&xs [0][xr][xc];
    const unsigned xlds1 = (unsigned)(size_t)&xs [1][xr][xc];
    const unsigned wlds0 = (unsigned)(size_t)&wsh[0][tid][0];
    const unsigned wlds1 = (unsigned)(size_t)&wsh[1][tid][0];

    // wave tiling: 8 waves = 2 (row blocks) x 4 (col blocks); each wave 16x64
    const int lane = tid & 31;
    const int w    = tid >> 5;
    const int wm   = w & 1;                 // 16-row block
    const int wn   = w >> 1;                // 64-col block (0..3)
    const int l15  = lane & 15;
    const int hiK  = (lane >> 4) << 1;      // lanes 16..31 hold K+2,K+3

    // issue chunk 0 into buffer 0 (9 async b128 ops per thread)
    {
        async_b128(xlds0, xg);
#pragma unroll
        for (int i = 0; i < 8; ++i) async_b128(wlds0 + i * 16, wg + i * 4);
    }

    v8f acc[4] = {};
    const int NCHUNK = H_DIM / KC;          // 64
    for (int c = 0; c < NCHUNK; ++c) {
        const int cur = c & 1;
        __syncthreads();                    // buffer cur^1 no longer being read
        if (c + 1 < NCHUNK) {               // stream next chunk while computing
            const int k0 = (c + 1) * KC;
            const unsigned xl = (cur == 0) ? xlds1 : xlds0;
            const unsigned wl = (cur == 0) ? wlds1 : wlds0;
            async_b128(xl, xg + k0);
#pragma unroll
            for (int i = 0; i < 8; ++i) async_b128(wl + i * 16, wg + k0 + i * 4);
            asm volatile("s_wait_asynccnt 0x9" ::: "memory");  // chunk c landed
        } else {
            asm volatile("s_wait_asynccnt 0x0" ::: "memory");
        }
        __syncthreads();                    // all waves' chunk-c data visible

#pragma unroll
        for (int kk = 0; kk < KC; kk += 4) {
            v2f a = *(const v2f*)&xs[cur][(wm << 4) + l15][kk + hiK];
#pragma unroll
            for (int t = 0; t < 4; ++t) {   // A fragment reused across 4 B tiles
                v2f b = *(const v2f*)&wsh[cur][(wn << 6) + (t << 4) + l15][kk + hiK];
                acc[t] = __builtin_amdgcn_wmma_f32_16x16x4_f32(
                    false, a, false, b, (short)0, acc[t], false, false);
            }
        }
    }

    // epilogue: softcap + fixed-max exp + per-row partial sums
    const int colW = vBase + (wn << 6) + l15;      // + 16*t per tile
#pragma unroll
    for (int r = 0; r < 8; ++r) {
        const int row = rowBase + (wm << 4) + r + ((lane >> 4) << 3);
        const int yt  = y[row];
        float sz = 0.0f, se = 0.0f;
#pragma unroll
        for (int t = 0; t < 4; ++t) {
            float z = acc[t][r];
            z = SOFTCAP * tanhf(z * (1.0f / SOFTCAP));
            sz += z;
            se += __expf(z - SOFTCAP);
            if (yt == colW + (t << 4)) zy[row] = z;    // unique across grid
        }
#pragma unroll
        for (int m = 1; m < 16; m <<= 1) {             // reduce over 16 N-lanes
            sz += __shfl_xor(sz, m, 32);
            se += __shfl_xor(se, m, 32);
        }
        if (l15 == 0) {
            atomicAdd(&S1[row], se);
            atomicAdd(&S2[row], sz);
        }
    }
}

// ---------------------------------------------------------------------------
// Kernel 2: finalize  mean_n [ lse - (1-eps) z_y - (eps/V) S2 + 1e-4 lse^2 ]
// ---------------------------------------------------------------------------
__global__ __launch_bounds__(256)
void lmhead_finalize(const float* __restrict__ S1, const float* __restrict__ S2,
                     const float* __restrict__ zy, const int* __restrict__ y,
                     float* __restrict__ out, int N) {
    __shared__ float ssum[256];
    __shared__ int   scnt[256];
    const int tid = threadIdx.x;
    float acc = 0.0f; int cnt = 0;
    for (int n = tid; n < N; n += 256) {
        if (y[n] != IGNORE_IDX) {
            float lse = SOFTCAP + logf(S1[n]);
            float ce  = lse - (1.0f - EPS_LS) * zy[n]
                            - (EPS_LS / (float)V_DIM) * S2[n];
            acc += ce + ZLOSS * lse * lse;
            cnt += 1;
        }
    }
    ssum[tid] = acc; scnt[tid] = cnt;
    __syncthreads();
    for (int s = 128; s > 0; s >>= 1) {
        if (tid < s) { ssum[tid] += ssum[tid + s]; scnt[tid] += scnt[tid + s]; }
        __syncthreads();
    }
    if (tid == 0) {
        int nv = scnt[0] > 0 ? scnt[0] : 1;
        out[0] = ssum[0] / (float)nv;
    }
}

// ---------------------------------------------------------------------------
extern "C" void kernel_launch(void* const* d_in, const int* in_sizes, int n_in,
                              void* d_out, int out_size, void* d_ws, size_t ws_size,
                              hipStream_t stream) {
    const float* x = (const float*)d_in[0];
    const float* W = (const float*)d_in[1];
    const int*   y = (const int*)  d_in[2];
    const int N = in_sizes[0] / H_DIM;

    float* S1 = (float*)d_ws;
    float* S2 = S1 + N;
    float* zy = S2 + N;

    zero_ws<<<(3 * N + 255) / 256, 256, 0, stream>>>(S1, 3 * N);

    dim3 grid(N / MT, V_DIM / NT);   // rows fastest -> W tile reuse in L2
    lmhead_partial<<<grid, 256, 0, stream>>>(x, W, y, S1, S2, zy);

    lmhead_finalize<<<1, 256, 0, stream>>>(S1, S2, zy, y, (float*)d_out, N);
}